// GCNLayer_28157805593244
// MI455X (gfx1250) — compile-verified
//
#include <hip/hip_runtime.h>
#include <stdint.h>

// GCN layer on MI455X (gfx1250):
//   phase 1: zero agg workspace (float4 stores)
//   phase 2: edge scatter-add  agg[dst] += node_feats[src]   (L2-resident f32 atomics)
//   phase 3: out = agg @ W^T + b via V_WMMA_F32_16X16X4_F32, LDS-staged tiles,
//            async global->LDS copy for the shared W tile.

#define IN_FEATS  128
#define OUT_FEATS 128
#define LDS_STRIDE 132   // 128 + 4 pad floats -> conflict-free ds_load_b64 fragment reads
#define USE_ASYNC_LDS 1

typedef float v2f __attribute__((ext_vector_type(2)));
typedef float v8f __attribute__((ext_vector_type(8)));

__global__ void gcn_zero_kernel(float4* __restrict__ agg, int n4) {
    int i = blockIdx.x * blockDim.x + threadIdx.x;
    if (i < n4) agg[i] = make_float4(0.f, 0.f, 0.f, 0.f);
}

// One wave per edge: 32 lanes x float4 = 128 feats, coalesced 512B gather,
// then 4 global_atomic_add_f32 per lane into agg[dst] (resolves in L2).
__global__ void gcn_scatter_kernel(const float* __restrict__ node_feats,
                                   const int*   __restrict__ src,
                                   const int*   __restrict__ dst,
                                   float*       __restrict__ agg,
                                   int n_edges) {
    int edge = (blockIdx.x * blockDim.x + threadIdx.x) >> 5;
    int lane = threadIdx.x & 31;
    if (edge >= n_edges) return;

    int s = src[edge];
    int d = dst[edge];

    float4 v = reinterpret_cast<const float4*>(node_feats + (size_t)s * IN_FEATS)[lane];
    float* o = agg + (size_t)d * IN_FEATS + lane * 4;
    atomicAdd(o + 0, v.x);
    atomicAdd(o + 1, v.y);
    atomicAdd(o + 2, v.z);
    atomicAdd(o + 3, v.w);
}

// Block = 128 threads = 4 waves. blockIdx.x -> n-tile (shared W tile in LDS),
// each wave owns one 16-row m-tile (private A tile in LDS).
// WMMA f32 16x16x4, K accumulated over 32 issues from LDS fragments.
//
// A-matrix lane layout (ISA 7.12.2): lane L<16 holds A[M=L][K=k0..k0+1],
// lanes 16-31 hold A[M=L-16][K=k0+2..k0+3] -> one 8-byte LDS read per lane.
// B[k][n] = W[n0+n][k0+k] -> identical fragment pattern from the W tile.
// C/D: VGPR r -> row m0 + r + 8*(lane>>4), col n0 + (lane&15).
__global__ void __launch_bounds__(128)
gcn_gemm_wmma_kernel(const float* __restrict__ agg,
                     const float* __restrict__ W,
                     const float* __restrict__ bias,
                     float*       __restrict__ out,
                     int n_nodes) {
    __shared__ float sW[16 * LDS_STRIDE];        // shared B tile (rows n0..n0+15 of W)
    __shared__ float sA[4][16 * LDS_STRIDE];     // per-wave A tile

    const int lane = threadIdx.x;            // 0..31
    const int wave = threadIdx.y;            // 0..3
    const int tid  = wave * 32 + lane;
    const int half = lane >> 4;
    const int l16  = lane & 15;

    const int n0 = blockIdx.x * 16;
    const int m0 = (blockIdx.y * 4 + wave) * 16;

    // ---- stage shared W tile: 16 rows x 128 floats, 512 float4, 4 per thread ----
    #pragma unroll
    for (int i = 0; i < 4; ++i) {
        const int t   = i * 128 + tid;
        const int row = t >> 5;              // 0..15
        const int c4  = t & 31;              // float4 column
        const float* gp = W + (size_t)(n0 + row) * IN_FEATS + c4 * 4;
        float*       lp = &sW[row * LDS_STRIDE + c4 * 4];
#if USE_ASYNC_LDS
        // ASYNCcnt-tracked global->LDS DMA (GLOBAL_LOAD_ASYNC_TO_LDS_B128).
        uint32_t ldsoff = (uint32_t)(uintptr_t)lp;   // low 32 bits of generic AS3 addr
        asm volatile("global_load_async_to_lds_b128 %0, %1, off"
                     :: "v"(ldsoff), "v"(gp) : "memory");
#else
        *(float4*)lp = *(const float4*)gp;
#endif
    }

    // ---- stage this wave's A tile: 16 rows x 128 floats, coalesced b128 ----
    // OOB rows clamped: A row M only feeds output row M, which is never stored
    // when OOB, so duplicated data is harmless. Uniform branch, no divergence.
    #pragma unroll
    for (int i = 0; i < 16; ++i) {
        int row = m0 + i;
        if (row >= n_nodes) row = n_nodes - 1;
        float4 v = *(const float4*)(agg + (size_t)row * IN_FEATS + lane * 4);
        *(float4*)&sA[wave][i * LDS_STRIDE + lane * 4] = v;
    }

#if USE_ASYNC_LDS
    asm volatile("s_wait_asynccnt 0x0" ::: "memory");
#endif
    __syncthreads();

    // ---- WMMA accumulation over K=128 from LDS fragments ----
    const float* aRow = &sA[wave][l16 * LDS_STRIDE + 2 * half];
    const float* bRow = &sW[l16 * LDS_STRIDE + 2 * half];

    v8f c = {};
    #pragma unroll
    for (int k0 = 0; k0 < IN_FEATS; k0 += 4) {
        v2f a = *(const v2f*)(aRow + k0);    // ds_load_b64, conflict-free
        v2f b = *(const v2f*)(bRow + k0);
        c = __builtin_amdgcn_wmma_f32_16x16x4_f32(
                /*neg_a=*/false, a, /*neg_b=*/false, b,
                /*c_mod=*/(short)0, c, /*reuse_a=*/false, /*reuse_b=*/false);
    }

    // ---- epilogue: bias + store ----
    const float bv = bias[n0 + l16];
    #pragma unroll
    for (int r = 0; r < 8; ++r) {
        const int row = m0 + r + 8 * half;
        if (row < n_nodes) {
            out[(size_t)row * OUT_FEATS + n0 + l16] = c[r] + bv;
        }
    }
}

extern "C" void kernel_launch(void* const* d_in, const int* in_sizes, int n_in,
                              void* d_out, int out_size, void* d_ws, size_t ws_size,
                              hipStream_t stream) {
    // setup_inputs order: node_feats, edge_feats, src, dst, W, b
    const float* node_feats = (const float*)d_in[0];
    // edge_feats (d_in[1]) is unused by the reference message function
    const int*   src        = (const int*)d_in[2];
    const int*   dst        = (const int*)d_in[3];
    const float* W          = (const float*)d_in[4];
    const float* bias       = (const float*)d_in[5];
    float*       out        = (float*)d_out;

    const int n_nodes = in_sizes[0] / IN_FEATS;
    const int n_edges = in_sizes[2];

    float* agg = (float*)d_ws;   // n_nodes * 128 floats (~25.6 MB), L2-resident

    // Phase 1: zero agg (float4 granularity; IN_FEATS divisible by 4).
    {
        int n4 = n_nodes * (IN_FEATS / 4);
        gcn_zero_kernel<<<(n4 + 255) / 256, 256, 0, stream>>>((float4*)agg, n4);
    }

    // Phase 2: scatter-add. 256 threads = 8 waves per block, 1 edge per wave.
    {
        int waves_per_block = 256 / 32;
        int blocks = (n_edges + waves_per_block - 1) / waves_per_block;
        gcn_scatter_kernel<<<blocks, 256, 0, stream>>>(node_feats, src, dst, agg, n_edges);
    }

    // Phase 3: WMMA GEMM + bias. 4 waves per block, one 16x16 tile per wave.
    {
        const int waves_per_block = 4;
        int m_tiles = (n_nodes + 15) / 16;
        dim3 block(32, waves_per_block);
        dim3 grid(OUT_FEATS / 16, (m_tiles + waves_per_block - 1) / waves_per_block);
        gcn_gemm_wmma_kernel<<<grid, block, 0, stream>>>(agg, W, bias, out, n_nodes);
    }
}